// GNNRegressor_584115552932
// MI455X (gfx1250) — compile-verified
//
#include <hip/hip_runtime.h>

// ---------------------------------------------------------------------------
// GCN (2x GCNConv + linear) for MI455X / gfx1250, wave32.
//   deg -> dinv -> [WMMA GEMM -> selfloop -> edge atomic agg] x2 -> final dot
// Dense GEMMs use V_WMMA_F32_16X16X4_F32 (f32 in, f32 acc) so the math
// matches the f32 reference; runtime is dominated by the HBM-bound
// gather/atomic-scatter over 3.2M edges (~2.5 GB -> ~110 us roofline).
// ---------------------------------------------------------------------------

typedef float v2f __attribute__((ext_vector_type(2)));
typedef float v8f __attribute__((ext_vector_type(8)));

// ---------------- degree / normalization -----------------------------------

__global__ void init_deg_kernel(float* __restrict__ deg, int n) {
  int i = blockIdx.x * blockDim.x + threadIdx.x;
  if (i < n) deg[i] = 1.0f;  // self loop
}

__global__ void deg_count_kernel(const int* __restrict__ dst,
                                 float* __restrict__ deg, int e) {
  int i = blockIdx.x * blockDim.x + threadIdx.x;
  if (i < e) atomicAdd(&deg[dst[i]], 1.0f);
}

__global__ void dinv_kernel(float* __restrict__ deg, int n) {
  int i = blockIdx.x * blockDim.x + threadIdx.x;
  if (i < n) deg[i] = rsqrtf(deg[i]);  // in place: deg -> deg^{-1/2}
}

// ---------------- dense GEMM via WMMA f32 16x16x4 ---------------------------
// One wave computes a 16 x NCOLS output tile. A tile staged in LDS with
// padded stride (bank-conflict free). preRelu applies relu(A + preBias)
// on load (fuses the previous layer's bias+relu into the next GEMM).

template <int K, int NCOLS>
__global__ __launch_bounds__(128) void gemm_wmma_kernel(
    const float* __restrict__ A, const float* __restrict__ W,
    const float* __restrict__ preBias, int preRelu,
    float* __restrict__ Cout, int M) {
  constexpr int NT  = NCOLS / 16;  // 16-wide output tiles
  constexpr int LDK = K + 1;       // padded LDS row stride
  __shared__ float tile[4 * 16 * LDK];  // 4 waves per block

  const int lane = threadIdx.x & 31;
  const int wib  = threadIdx.x >> 5;
  const int row0 = (blockIdx.x * 4 + wib) * 16;
  float* t = tile + wib * 16 * LDK;

  // Stage 16 x K A-tile into LDS (coalesced; clamp rows past M).
  for (int i = lane; i < 16 * K; i += 32) {
    int r = i / K, c = i % K;
    int gr = row0 + r;
    if (gr >= M) gr = M - 1;
    float v = A[(long)gr * K + c];
    if (preRelu) v = fmaxf(v + preBias[c], 0.0f);
    t[r * LDK + c] = v;
  }
  __syncthreads();

  v8f acc[NT] = {};
  // A 16x4 f32 fragment: lanes 0-15 -> K = k0+0,k0+1 ; lanes 16-31 -> k0+2,k0+3
  const int rA  = lane & 15;
  const int kk  = (lane >> 4) * 2;
  const int col = lane & 15;

  for (int k0 = 0; k0 < K; k0 += 4) {
    v2f a;
    a.x = t[rA * LDK + k0 + kk + 0];
    a.y = t[rA * LDK + k0 + kk + 1];
#pragma unroll
    for (int n = 0; n < NT; ++n) {
      v2f b;  // B 4x16 fragment, mirrored layout (W row-major [K][NCOLS])
      b.x = W[(k0 + kk + 0) * NCOLS + n * 16 + col];
      b.y = W[(k0 + kk + 1) * NCOLS + n * 16 + col];
      acc[n] = __builtin_amdgcn_wmma_f32_16x16x4_f32(
          false, a, false, b, (short)0, acc[n], false, false);
    }
  }

  // C/D layout: VGPR i -> row i (lanes 0-15) / row i+8 (lanes 16-31).
  // Tile-fullness is wave-uniform: hoist it into a single scalar branch so
  // the common path is NT*8 unguarded b32 stores (no per-store exec dance).
  const int rowOff = (lane >> 4) * 8;
  const int row0s  = __builtin_amdgcn_readfirstlane(row0);
  if (row0s + 16 <= M) {
#pragma unroll
    for (int n = 0; n < NT; ++n) {
#pragma unroll
      for (int i = 0; i < 8; ++i) {
        Cout[(long)(row0s + rowOff + i) * NCOLS + n * 16 + col] = acc[n][i];
      }
    }
  } else {
#pragma unroll
    for (int n = 0; n < NT; ++n) {
#pragma unroll
      for (int i = 0; i < 8; ++i) {
        int row = row0s + rowOff + i;
        if (row < M) Cout[(long)row * NCOLS + n * 16 + col] = acc[n][i];
      }
    }
  }
}

// ---------------- self-loop term: agg = dinv^2 * pre (also initializes agg) -
// float4 vectorized: b128 load + b128 store per thread.

template <int C>
__global__ void selfloop_kernel(const float* __restrict__ pre,
                                const float* __restrict__ dinv,
                                float* __restrict__ agg, int n) {
  constexpr int G = C / 4;
  int i = blockIdx.x * blockDim.x + threadIdx.x;  // one float4 per thread
  if (i >= n * G) return;
  int node = i / G;
  float w = dinv[node];
  float w2 = w * w;
  float4 v = ((const float4*)pre)[i];
  v.x *= w2; v.y *= w2; v.z *= w2; v.w *= w2;
  ((float4*)agg)[i] = v;
}

// ---------------- edge aggregation: agg[dst] += dinv[s]*dinv[d]*pre[src] ----
// C/4 threads per edge; float4 gather, 4x global_atomic_add_f32 scatter.

template <int C>
__global__ void edge_agg_kernel(const float* __restrict__ pre,
                                const int* __restrict__ src,
                                const int* __restrict__ dst,
                                const float* __restrict__ dinv,
                                float* __restrict__ agg, int e) {
  constexpr int G = C / 4;
  int tid = blockIdx.x * blockDim.x + threadIdx.x;
  int eIdx = tid / G;
  int g    = tid % G;
  if (eIdx >= e) return;
  int s = src[eIdx], d = dst[eIdx];
  float w = dinv[s] * dinv[d];
  const float4 v = *(const float4*)(pre + (long)s * C + g * 4);
  float* p = agg + (long)d * C + g * 4;
  atomicAdd(p + 0, w * v.x);
  atomicAdd(p + 1, w * v.y);
  atomicAdd(p + 2, w * v.z);
  atomicAdd(p + 3, w * v.w);
}

// ---------------- final: out = relu(h2 + b2) @ Wl + bl (wave per node) ------

__global__ void final_kernel(const float* __restrict__ h2,
                             const float* __restrict__ b2,
                             const float* __restrict__ Wl,
                             const float* __restrict__ bl,
                             float* __restrict__ out, int n) {
  int gid  = blockIdx.x * blockDim.x + threadIdx.x;
  int node = gid >> 5;
  int lane = gid & 31;
  if (node >= n) return;
  float v = fmaxf(h2[(long)node * 32 + lane] + b2[lane], 0.0f) * Wl[lane];
#pragma unroll
  for (int off = 16; off > 0; off >>= 1) v += __shfl_xor(v, off, 32);
  if (lane == 0) out[node] = v + bl[0];
}

// ---------------------------------------------------------------------------

extern "C" void kernel_launch(void* const* d_in, const int* in_sizes, int n_in,
                              void* d_out, int out_size, void* d_ws,
                              size_t ws_size, hipStream_t stream) {
  const float* x  = (const float*)d_in[0];
  const int*   ei = (const int*)d_in[1];
  const float* W1 = (const float*)d_in[2];
  const float* b1 = (const float*)d_in[3];
  const float* W2 = (const float*)d_in[4];
  const float* b2 = (const float*)d_in[5];
  const float* Wl = (const float*)d_in[6];
  const float* bl = (const float*)d_in[7];
  float* out = (float*)d_out;

  const int N = in_sizes[0] / 128;  // 100000
  const int E = in_sizes[1] / 2;    // 3200000
  const int* src  = ei;
  const int* dstp = ei + E;

  // Workspace layout (floats): deg/dinv | h1pre | h1agg | h2pre | h2agg
  float* ws    = (float*)d_ws;
  float* deg   = ws;
  float* h1pre = deg + (size_t)N;
  float* h1agg = h1pre + (size_t)N * 64;
  float* h2pre = h1agg + (size_t)N * 64;
  float* h2agg = h2pre + (size_t)N * 32;
  (void)ws_size; (void)n_in; (void)out_size;

  const int T = 256;
  // --- normalization ---
  init_deg_kernel<<<(N + T - 1) / T, T, 0, stream>>>(deg, N);
  deg_count_kernel<<<(E + T - 1) / T, T, 0, stream>>>(dstp, deg, E);
  dinv_kernel<<<(N + T - 1) / T, T, 0, stream>>>(deg, N);

  const int tiles = (N + 15) / 16;  // 16-row tiles, 4 waves/block
  // --- layer 1: h1pre = x @ W1 ; agg with norm ---
  gemm_wmma_kernel<128, 64><<<(tiles + 3) / 4, 128, 0, stream>>>(
      x, W1, nullptr, 0, h1pre, N);
  selfloop_kernel<64><<<(N * 16 + T - 1) / T, T, 0, stream>>>(h1pre, deg,
                                                              h1agg, N);
  edge_agg_kernel<64><<<(E * 16 + T - 1) / T, T, 0, stream>>>(
      h1pre, src, dstp, deg, h1agg, E);

  // --- layer 2: h2pre = relu(h1agg + b1) @ W2 ; agg with norm ---
  gemm_wmma_kernel<64, 32><<<(tiles + 3) / 4, 128, 0, stream>>>(
      h1agg, W2, b1, 1, h2pre, N);
  selfloop_kernel<32><<<(N * 8 + T - 1) / T, T, 0, stream>>>(h2pre, deg,
                                                             h2agg, N);
  edge_agg_kernel<32><<<(E * 8 + T - 1) / T, T, 0, stream>>>(
      h2pre, src, dstp, deg, h2agg, E);

  // --- head: out = relu(h2agg + b2) @ Wl + bl ---
  final_kernel<<<(N * 32 + T - 1) / T, T, 0, stream>>>(h2agg, b2, Wl, bl, out,
                                                       N);
}